// VGAELinkPredictor_36464272343629
// MI455X (gfx1250) — compile-verified
//
#include <hip/hip_runtime.h>
#include <hip/hip_bf16.h>
#include <math.h>

typedef __attribute__((ext_vector_type(2))) float v2f;
typedef __attribute__((ext_vector_type(8))) float v8f;

#define CIN   128   // input channels
#define COUT  64    // output channels per head (mu / logstd)
#define MAX_LOGSTD 10.0f

// ---------------------------------------------------------------------------
// 1) init: zero the aggregation accumulator, set deg = 1 (self loop)
// ---------------------------------------------------------------------------
__global__ void vgae_init(float* __restrict__ agg, unsigned int* __restrict__ deg,
                          long long n128, int N) {
    long long i = (long long)blockIdx.x * blockDim.x + threadIdx.x;
    if (i < n128) agg[i] = 0.0f;
    if (i < N)    deg[i] = 1u;           // self-loop contributes 1 to degree
}

// ---------------------------------------------------------------------------
// 2) degree histogram over dst (u32 hardware atomics)
// ---------------------------------------------------------------------------
__global__ void vgae_degree(const long long* __restrict__ dst,
                            unsigned int* __restrict__ deg, long long E) {
    long long i = (long long)blockIdx.x * blockDim.x + threadIdx.x;
    if (i < E) atomicAdd(&deg[(int)dst[i]], 1u);
}

// ---------------------------------------------------------------------------
// 3) dinv = rsqrt(deg)   (written in place over the u32 degree buffer)
// ---------------------------------------------------------------------------
__global__ void vgae_dinv(unsigned int* __restrict__ degi,
                          float* __restrict__ dinv, int N) {
    int i = blockIdx.x * blockDim.x + threadIdx.x;
    if (i < N) {
        float d = (float)degi[i];        // deg >= 1 always (self-loop)
        dinv[i] = rsqrtf(d);
    }
}

// ---------------------------------------------------------------------------
// 4) h = x @ [W_mu | W_ls]  via V_WMMA_F32_16X16X4_F32 (fp32-exact)
//    block = 256 threads (8 waves). Block owns 16 rows of x (staged in LDS,
//    shared by all 8 waves); wave w owns output cols [16w, 16w+16).
//    waves 0-3 -> W_mu (h cols 0..63), waves 4-7 -> W_ls (h cols 64..127).
// ---------------------------------------------------------------------------
__global__ void __launch_bounds__(256)
vgae_gemm_wmma(const float* __restrict__ x,
               const float* __restrict__ Wmu,
               const float* __restrict__ Wls,
               float* __restrict__ h, int N) {
    __shared__ float xs[16][CIN + 4];    // +4 pad: conflict-free A reads
    const int tid  = threadIdx.x;
    const int wave = tid >> 5;           // 0..7 -> 16-col tile
    const int lane = tid & 31;
    const long long row0 = (long long)blockIdx.x * 16;

    // Cooperative stage of the 16x128 x-tile (float4, 2 iters/thread)
    for (int i = tid; i < 16 * (CIN / 4); i += 256) {
        int r  = i >> 5;                 // CIN/4 == 32
        int c4 = i & 31;
        float4 v = make_float4(0.f, 0.f, 0.f, 0.f);
        if (row0 + r < N)
            v = ((const float4*)(x + (row0 + r) * CIN))[c4];
        xs[r][c4 * 4 + 0] = v.x;  xs[r][c4 * 4 + 1] = v.y;
        xs[r][c4 * 4 + 2] = v.z;  xs[r][c4 * 4 + 3] = v.w;
    }
    __syncthreads();

    // Fragment coordinates per ISA 7.12.2 (wave32):
    //   A 16x4 f32: lane -> row m = lane&15; lanes 16-31 hold K+2.
    //   B 4x16 f32: lane -> col n = lane&15; lanes 16-31 hold K+2 (mirror of A).
    //   D 16x16  : VGPR r -> row r (lanes 0-15) / row r+8 (lanes 16-31).
    const int m  = lane & 15;
    const int kg = (lane >> 4) << 1;     // 0 or 2
    const int n  = lane & 15;
    const float* W = (wave < 4) ? Wmu : Wls;
    const int wc = ((wave & 3) << 4) + n;        // column 0..63 inside W

    v8f acc = {0.f, 0.f, 0.f, 0.f, 0.f, 0.f, 0.f, 0.f};
    #pragma unroll
    for (int k0 = 0; k0 < CIN; k0 += 4) {
        v2f a, b;
        a.x = xs[m][k0 + kg + 0];
        a.y = xs[m][k0 + kg + 1];
        b.x = W[(k0 + kg + 0) * COUT + wc];      // W is [128,64] row-major
        b.y = W[(k0 + kg + 1) * COUT + wc];
        acc = __builtin_amdgcn_wmma_f32_16x16x4_f32(
                  /*neg_a=*/false, a, /*neg_b=*/false, b,
                  /*c_mod=*/(short)0, acc, /*reuse_a=*/false, /*reuse_b=*/false);
    }

    const int hcol  = (wave << 4) + n;           // 0..127 in combined h
    const int rbase = (lane >> 4) << 3;          // 0 or 8
    #pragma unroll
    for (int r = 0; r < 8; ++r) {
        long long row = row0 + rbase + r;
        if (row < N) h[row * CIN + hcol] = acc[r];
    }
}

// ---------------------------------------------------------------------------
// 5) edge aggregation: one wave per edge (incl. N implicit self-loops).
//    float4 gather of h[src] (L2-resident), scaled hardware f32 atomics into
//    agg[dst]. 32 lanes x 4 ch = 128 channels.
// ---------------------------------------------------------------------------
__global__ void __launch_bounds__(256)
vgae_aggregate(const long long* __restrict__ src,
               const long long* __restrict__ dst,
               const float* __restrict__ dinv,
               const float* __restrict__ h,
               float* __restrict__ agg,
               long long E, int N) {
    long long t = (long long)blockIdx.x * (blockDim.x >> 5) + (threadIdx.x >> 5);
    const int lane = threadIdx.x & 31;
    if (t >= E + (long long)N) return;

    int s, d;
    if (t < E) { s = (int)src[t]; d = (int)dst[t]; }
    else       { s = d = (int)(t - E); }         // self-loop

    const float norm = dinv[s] * dinv[d];
    const float4 v = ((const float4*)(h + (long long)s * CIN))[lane];
    float* o = agg + (long long)d * CIN + lane * 4;
    unsafeAtomicAdd(o + 0, v.x * norm);          // global_atomic_add_f32
    unsafeAtomicAdd(o + 1, v.y * norm);
    unsafeAtomicAdd(o + 2, v.z * norm);
    unsafeAtomicAdd(o + 3, v.w * norm);
}

// ---------------------------------------------------------------------------
// 6) finalize: bias, clamp logstd, reparameterize
// ---------------------------------------------------------------------------
__global__ void vgae_finalize(const float* __restrict__ agg,
                              const float* __restrict__ b_mu,
                              const float* __restrict__ b_ls,
                              const float* __restrict__ eps,
                              float* __restrict__ z, long long n64) {
    long long i = (long long)blockIdx.x * blockDim.x + threadIdx.x;
    if (i >= n64) return;
    const int c      = (int)(i & (COUT - 1));
    const long long n = i >> 6;
    const float mu = agg[n * CIN + c] + b_mu[c];
    float ls = agg[n * CIN + COUT + c] + b_ls[c];
    ls = fminf(ls, MAX_LOGSTD);
    z[i] = mu + eps[i] * expf(ls);
}

// ---------------------------------------------------------------------------
extern "C" void kernel_launch(void* const* d_in, const int* in_sizes, int n_in,
                              void* d_out, int out_size, void* d_ws, size_t ws_size,
                              hipStream_t stream) {
    const float*     x    = (const float*)d_in[0];
    const long long* edge = (const long long*)d_in[1];
    const float*     Wmu  = (const float*)d_in[2];
    const float*     bmu  = (const float*)d_in[3];
    const float*     Wls  = (const float*)d_in[4];
    const float*     bls  = (const float*)d_in[5];
    const float*     eps  = (const float*)d_in[6];
    float*           z    = (float*)d_out;

    const int       N = in_sizes[0] / CIN;
    const long long E = (long long)in_sizes[1] / 2;
    const long long* srcv = edge;
    const long long* dstv = edge + E;

    // workspace: h[N*128] | agg[N*128] | deg[N]  (~103 MB; h+agg sit in L2)
    float*        h    = (float*)d_ws;
    float*        agg  = h + (size_t)N * CIN;
    unsigned int* deg  = (unsigned int*)(agg + (size_t)N * CIN);
    float*        dinv = (float*)deg;

    const int thr = 256;
    const long long n128 = (long long)N * CIN;
    const long long n64  = (long long)N * COUT;
    const long long T    = E + (long long)N;

    vgae_init     <<<(unsigned)((n128 + thr - 1) / thr), thr, 0, stream>>>(agg, deg, n128, N);
    vgae_degree   <<<(unsigned)((E + thr - 1) / thr),   thr, 0, stream>>>(dstv, deg, E);
    vgae_dinv     <<<(unsigned)((N + thr - 1) / thr),   thr, 0, stream>>>(deg, dinv, N);
    vgae_gemm_wmma<<<(unsigned)((N + 15) / 16),         256, 0, stream>>>(x, Wmu, Wls, h, N);
    vgae_aggregate<<<(unsigned)((T + 7) / 8),           256, 0, stream>>>(srcv, dstv, dinv, h, agg, E, N);
    vgae_finalize <<<(unsigned)((n64 + thr - 1) / thr), thr, 0, stream>>>(agg, bmu, bls, eps, z, n64);
}